// LocalNetwork_29197187678704
// MI455X (gfx1250) — compile-verified
//
#include <hip/hip_runtime.h>
#include <math.h>

typedef __attribute__((ext_vector_type(2))) float v2f;
typedef __attribute__((ext_vector_type(8))) float v8f;

#define B_DIM   512
#define C_DIM   4
#define H_DIM   125
#define W_DIM   125
#define K_DIM   62500      // C*H*W
#define N_DIM   20
#define HW_DIM  (H_DIM * W_DIM)

#define M_TILES   (B_DIM / 16)   // 32
#define N_TILES   2              // ceil(20/16)
#define KCHUNKS   25
#define KCHUNK    (K_DIM / KCHUNKS)   // 2500; 625 k-steps of 4 = 125 x unroll-5
#define NUM_TILES (M_TILES * N_TILES) // 64

// ---------------------------------------------------------------------------
// Phase 1a: split-K partial GEMM via V_WMMA_F32_16X16X4_F32.
// Grid = 64 tiles x 25 K-chunks = 1600 wave32 workgroups.
// A layout (16x4 f32): lanes 0-15 -> M=lane, K={0,1} in v0/v1; lanes 16-31 -> K={2,3}.
// B layout (4x16 f32): lanes 0-15 -> N=lane, K={0,1}; lanes 16-31 -> K={2,3}.
// D layout (16x16 f32): VGPR r, lanes 0-15 -> row r, lanes 16-31 -> row r+8.
// Inner loop uses running pointers + constant offsets so the unrolled body is
// immediate-offset loads + dual-issued mask-mul + v_wmma, no address mads.
// ---------------------------------------------------------------------------
__global__ __launch_bounds__(32)
void stn_gemm_partial(const float* __restrict__ img,
                      const float* __restrict__ W1,
                      float* __restrict__ partial) {
  const int wid   = blockIdx.x;            // 0..1599
  const int kc    = wid % KCHUNKS;
  const int tile  = wid / KCHUNKS;         // 0..63
  const int mTile = tile >> 1;             // 0..31
  const int nTile = tile & 1;              // 0..1
  const int lane  = threadIdx.x;           // 0..31 (one wave32, EXEC all ones)
  const int half  = lane >> 4;             // 0/1 -> K offset 0 / 2
  const int l16   = lane & 15;

  const int m = mTile * 16 + l16;          // A row handled by this lane
  const int n = nTile * 16 + l16;          // B column handled by this lane
  const bool  nvalid = (n < N_DIM);
  const float bmask  = nvalid ? 1.0f : 0.0f;

  const int k0 = kc * KCHUNK;
  const float* __restrict__ Aptr = img + (size_t)m * K_DIM + k0 + half * 2;
  const float* __restrict__ Bptr =
      W1 + (nvalid ? n : 0) + (size_t)(k0 + half * 2) * N_DIM;

  v8f acc = {};
#pragma unroll 5
  for (int it = 0; it < KCHUNK / 4; ++it) {
    v2f a, b;
    a.x = Aptr[0];
    a.y = Aptr[1];
    b.x = Bptr[0] * bmask;
    b.y = Bptr[N_DIM] * bmask;
    acc = __builtin_amdgcn_wmma_f32_16x16x4_f32(
        /*neg_a=*/false, a, /*neg_b=*/false, b,
        /*c_mod=*/(short)0, acc, /*reuse_a=*/false, /*reuse_b=*/false);
    Aptr += 4;
    Bptr += 4 * N_DIM;
  }

  // Store partial tile, linearized as elem = row_in_tile*16 + col_in_tile.
  float* __restrict__ pt = partial + (size_t)wid * 256;
#pragma unroll
  for (int r = 0; r < 8; ++r) {
    pt[(r + half * 8) * 16 + l16] = acc[r];
  }
}

// ---------------------------------------------------------------------------
// Phase 1b: deterministic split-K reduction (fixed summation order) + bias +
// tanh -> H(512, 20).
// ---------------------------------------------------------------------------
__global__ __launch_bounds__(256)
void stn_reduce_tanh(const float* __restrict__ partial,
                     const float* __restrict__ b1,
                     float* __restrict__ Hout) {
  const int idx = blockIdx.x * blockDim.x + threadIdx.x;   // 0..16383
  if (idx >= NUM_TILES * 256) return;
  const int tile = idx >> 8;          // 0..63
  const int e    = idx & 255;         // element within 16x16 tile
  const int mTile = tile >> 1;
  const int nTile = tile & 1;
  const int row = mTile * 16 + (e >> 4);
  const int col = nTile * 16 + (e & 15);

  const float* __restrict__ p = partial + (size_t)tile * KCHUNKS * 256 + e;
  float sum = 0.0f;
#pragma unroll
  for (int kc = 0; kc < KCHUNKS; ++kc) sum += p[(size_t)kc * 256];

  if (col < N_DIM) {
    Hout[row * N_DIM + col] = tanhf(sum + b1[col]);
  }
}

// ---------------------------------------------------------------------------
// Phase 2: theta = sigmoid(H @ W2 + b2) * 2pi; emit Mat (B,2,3) and (cos,sin)
// ---------------------------------------------------------------------------
__global__ void stn_head(const float* __restrict__ Hin,
                         const float* __restrict__ W2,
                         const float* __restrict__ b2,
                         float* __restrict__ Mat_out,
                         float* __restrict__ cs_ws) {
  const int b = blockIdx.x * blockDim.x + threadIdx.x;
  if (b >= B_DIM) return;
  float acc = b2[0];
#pragma unroll
  for (int j = 0; j < N_DIM; ++j) acc += Hin[b * N_DIM + j] * W2[j];
  const float sig   = 1.0f / (1.0f + expf(-acc));
  const float theta = sig * 6.28318530717958647692f;
  const float c = cosf(theta);
  const float s = sinf(theta);
  Mat_out[b * 6 + 0] = c;
  Mat_out[b * 6 + 1] = -s;
  Mat_out[b * 6 + 2] = 0.0f;
  Mat_out[b * 6 + 3] = s;
  Mat_out[b * 6 + 4] = c;
  Mat_out[b * 6 + 5] = 0.0f;
  cs_ws[b * 2 + 0] = c;
  cs_ws[b * 2 + 1] = s;
}

// ---------------------------------------------------------------------------
// Phase 3: affine_grid (pure rotation, align_corners=True) + bilinear
// grid_sample with zero padding. One thread per (b,y,x); 4 channels reuse the
// same weights. Whole img (128MB) fits the 192MB L2 so the 16 gather taps hit
// cache after first touch; output writes are coalesced in x. This stage is
// the ~256MB bandwidth floor of the whole op (~11us at 23.3 TB/s).
// ---------------------------------------------------------------------------
__global__ __launch_bounds__(256)
void stn_sample(const float* __restrict__ img,
                const float* __restrict__ cs,
                float* __restrict__ out) {
  const long long idx = (long long)blockIdx.x * blockDim.x + threadIdx.x;
  if (idx >= (long long)B_DIM * HW_DIM) return;
  const int b   = (int)(idx / HW_DIM);
  const int rem = (int)(idx % HW_DIM);
  const int y   = rem / W_DIM;
  const int x   = rem % W_DIM;

  const float c = cs[2 * b + 0];
  const float s = cs[2 * b + 1];

  const float gx = -1.0f + 2.0f * (float)x / (float)(W_DIM - 1);
  const float gy = -1.0f + 2.0f * (float)y / (float)(H_DIM - 1);

  // grid = Mat @ (gx, gy, 1): x' = c*gx - s*gy ; y' = s*gx + c*gy
  const float xr = c * gx - s * gy;
  const float yr = s * gx + c * gy;

  const float ix = (xr + 1.0f) * 0.5f * (float)(W_DIM - 1);
  const float iy = (yr + 1.0f) * 0.5f * (float)(H_DIM - 1);

  const float ix0f = floorf(ix), iy0f = floorf(iy);
  const int ix0 = (int)ix0f, iy0 = (int)iy0f;
  const int ix1 = ix0 + 1,   iy1 = iy0 + 1;

  const float wx1 = ix - ix0f, wx0 = 1.0f - wx1;
  const float wy1 = iy - iy0f, wy0 = 1.0f - wy1;

  const bool vx0 = (ix0 >= 0) && (ix0 <= W_DIM - 1);
  const bool vx1 = (ix1 >= 0) && (ix1 <= W_DIM - 1);
  const bool vy0 = (iy0 >= 0) && (iy0 <= H_DIM - 1);
  const bool vy1 = (iy1 >= 0) && (iy1 <= H_DIM - 1);

  const int cx0 = min(max(ix0, 0), W_DIM - 1);
  const int cx1 = min(max(ix1, 0), W_DIM - 1);
  const int cy0 = min(max(iy0, 0), H_DIM - 1);
  const int cy1 = min(max(iy1, 0), H_DIM - 1);

  const float w00 = wx0 * wy0 * ((vx0 && vy0) ? 1.0f : 0.0f);
  const float w10 = wx1 * wy0 * ((vx1 && vy0) ? 1.0f : 0.0f);
  const float w01 = wx0 * wy1 * ((vx0 && vy1) ? 1.0f : 0.0f);
  const float w11 = wx1 * wy1 * ((vx1 && vy1) ? 1.0f : 0.0f);

  const float* __restrict__ ib = img + (size_t)b * C_DIM * HW_DIM;
#pragma unroll
  for (int ch = 0; ch < C_DIM; ++ch) {
    const float* __restrict__ p = ib + (size_t)ch * HW_DIM;
    const float v00 = p[cy0 * W_DIM + cx0];
    const float v10 = p[cy0 * W_DIM + cx1];
    const float v01 = p[cy1 * W_DIM + cx0];
    const float v11 = p[cy1 * W_DIM + cx1];
    out[((size_t)b * C_DIM + ch) * HW_DIM + rem] =
        v00 * w00 + v10 * w10 + v01 * w01 + v11 * w11;
  }
}

extern "C" void kernel_launch(void* const* d_in, const int* in_sizes, int n_in,
                              void* d_out, int out_size, void* d_ws, size_t ws_size,
                              hipStream_t stream) {
  (void)in_sizes; (void)n_in; (void)out_size; (void)ws_size;
  const float* img = (const float*)d_in[0];
  const float* W1  = (const float*)d_in[1];
  const float* b1  = (const float*)d_in[2];
  const float* W2  = (const float*)d_in[3];
  const float* b2  = (const float*)d_in[4];

  float* out_img = (float*)d_out;                                   // (B,C,H,W)
  float* out_mat = out_img + (size_t)B_DIM * C_DIM * HW_DIM;        // (B,2,3)

  // Workspace layout: [partials 1.6MB][H 40KB][cos/sin 4KB]
  float* ws_partial = (float*)d_ws;                                   // 64*25*256 f32
  float* ws_H       = ws_partial + (size_t)NUM_TILES * KCHUNKS * 256; // (512, 20)
  float* ws_cs      = ws_H + (size_t)B_DIM * N_DIM;                   // (512, 2)

  // Phase 1a: split-K WMMA partial GEMM: 1600 wave32 workgroups.
  stn_gemm_partial<<<dim3(NUM_TILES * KCHUNKS), dim3(32), 0, stream>>>(
      img, W1, ws_partial);

  // Phase 1b: deterministic reduction + bias + tanh.
  stn_reduce_tanh<<<dim3((NUM_TILES * 256) / 256), dim3(256), 0, stream>>>(
      ws_partial, b1, ws_H);

  // Phase 2: tiny head -> Mat + (cos, sin)
  stn_head<<<dim3(2), dim3(256), 0, stream>>>(ws_H, W2, b2, out_mat, ws_cs);

  // Phase 3: rotate + bilinear sample. One thread per (b, y, x).
  const long long npix = (long long)B_DIM * HW_DIM;
  const int blocks = (int)((npix + 255) / 256);
  stn_sample<<<dim3(blocks), dim3(256), 0, stream>>>(img, ws_cs, out_img);
}